// RoutesEncoder_14044543058415
// MI455X (gfx1250) — compile-verified
//
#include <hip/hip_runtime.h>
#include <math.h>

#define N_LOC     100000
#define D_IN      128
#define N_ROUTES  8192
#define ROUTE_LEN 100
#define D_OUT     256
#define LN_EPS    1e-5f

typedef __attribute__((ext_vector_type(2))) float v2f;
typedef __attribute__((ext_vector_type(4))) float v4f;
typedef __attribute__((ext_vector_type(8))) float v8f;

// -------------------------------------------------------------------------
// Kernel 1: gather + max-pool.  8 waves per block, one wave per route.
// Each lane owns dims [4*lane, 4*lane+4): one global_load_b128 per route
// step covers the whole 512B row per wave. Indices staged once into LDS.
// -------------------------------------------------------------------------
#define ROUTES_PER_BLOCK 8

__global__ void __launch_bounds__(256)
gather_maxpool_kernel(const float* __restrict__ emb,
                      const long long* __restrict__ idx,
                      float* __restrict__ pooled) {
    __shared__ int sIdx[ROUTES_PER_BLOCK * ROUTE_LEN];   // 800 ints

    const int tid  = threadIdx.x;
    const int wave = tid >> 5;
    const int lane = tid & 31;
    const int route = blockIdx.x * ROUTES_PER_BLOCK + wave;

    // stage all 8 routes' int64 indices -> int LDS
    const long long* gIdx = idx + (size_t)blockIdx.x * ROUTES_PER_BLOCK * ROUTE_LEN;
    for (int i = tid; i < ROUTES_PER_BLOCK * ROUTE_LEN; i += 256)
        sIdx[i] = (int)gIdx[i];
    __syncthreads();

    const int* myIdx = sIdx + wave * ROUTE_LEN;
    v4f m = { -3.402823466e38f, -3.402823466e38f,
              -3.402823466e38f, -3.402823466e38f };
#pragma unroll 4
    for (int j = 0; j < ROUTE_LEN; ++j) {
        const int loc = myIdx[j];
        const v4f v = *(const v4f*)(emb + (size_t)loc * D_IN + lane * 4);
        m.x = fmaxf(m.x, v.x);
        m.y = fmaxf(m.y, v.y);
        m.z = fmaxf(m.z, v.z);
        m.w = fmaxf(m.w, v.w);
    }
    *(v4f*)(pooled + (size_t)route * D_IN + lane * 4) = m;
}

// -------------------------------------------------------------------------
// Kernel 2: fused  pooled @ W + b  ->  LayerNorm(256)  ->  ReLU.
// Grid 512 (M/16), block 512 = 16 waves. Wave w computes the 16x16 tile at
// columns [16w,16w+16) with V_WMMA_F32_16X16X4_F32 over K=128 (32 steps).
// A strip staged in LDS (stride 132 words, bank-conflict-free); h tile goes
// through LDS (stride 260), then each wave LayerNorms one row.
// -------------------------------------------------------------------------
#define A_STRIDE 132
#define H_STRIDE 260

__global__ void __launch_bounds__(512)
gemm_ln_relu_kernel(const float* __restrict__ pooled,
                    const float* __restrict__ W,
                    const float* __restrict__ bias,
                    const float* __restrict__ gamma,
                    const float* __restrict__ beta,
                    float* __restrict__ out) {
    __shared__ float aT[16 * A_STRIDE];     // 16 rows x 128 cols of pooled
    __shared__ float hbuf[16 * H_STRIDE];   // 16 rows x 256 cols of h

    const int tid   = threadIdx.x;
    const int wave  = tid >> 5;             // 0..15 -> N tile
    const int lane  = tid & 31;
    const int half  = lane >> 4;            // 0 or 1
    const int l16   = lane & 15;
    const int rBase = blockIdx.x * 16;      // M strip origin
    const int col   = wave * 16 + l16;      // output column of this lane

    // ---- stage A strip: one float4 per thread (16B-aligned: 132*4=528) ----
    {
        const int r = tid >> 5;             // 0..15
        const int c = (tid & 31) * 4;       // 0..124
        *(v4f*)(aT + r * A_STRIDE + c) =
            *(const v4f*)(pooled + ((size_t)(rBase + r)) * D_IN + c);
    }
    __syncthreads();

    // ---- 32 x V_WMMA_F32_16X16X4_F32 ----
    // A layout (16x4 f32): lanes 0-15 hold row M=l16, K = k0 + {0,1};
    //                      lanes 16-31 hold row M=l16, K = k0 + {2,3}.
    // B layout (4x16 f32): mirrored: this lane supplies column `col`,
    //                      rows K = k0 + 2*half + {0,1}.
    v8f acc = {};
    const float* Wc = W + col;
#pragma unroll
    for (int k0 = 0; k0 < D_IN; k0 += 4) {
        const int ka = k0 + 2 * half;
        const v2f a = *(const v2f*)(aT + l16 * A_STRIDE + ka);   // 8B-aligned
        v2f b;
        b.x = Wc[(size_t)ka * D_OUT];
        b.y = Wc[(size_t)(ka + 1) * D_OUT];
        acc = __builtin_amdgcn_wmma_f32_16x16x4_f32(
            /*neg_a=*/false, a, /*neg_b=*/false, b,
            /*c_mod=*/(short)0, acc, /*reuse_a=*/false, /*reuse_b=*/false);
    }

    // ---- scatter C tile (+bias) into LDS ----
    // C/D layout: VGPR r -> M=r (lanes 0-15), M=r+8 (lanes 16-31); N = lane%16.
    const float bcol = bias[col];
#pragma unroll
    for (int r = 0; r < 8; ++r) {
        const int m = r + half * 8;
        hbuf[m * H_STRIDE + col] = acc[r] + bcol;
    }
    __syncthreads();

    // ---- LayerNorm + ReLU: wave `wave` handles row `wave` ----
    const int row = wave;
    float vals[8];
    float s = 0.f, s2 = 0.f;
#pragma unroll
    for (int j = 0; j < 8; ++j) {
        const float v = hbuf[row * H_STRIDE + lane + 32 * j];    // distinct banks
        vals[j] = v;
        s  += v;
        s2 += v * v;
    }
#pragma unroll
    for (int off = 16; off > 0; off >>= 1) {
        s  += __shfl_xor(s,  off, 32);
        s2 += __shfl_xor(s2, off, 32);
    }
    const float mean = s  * (1.f / D_OUT);
    const float var  = s2 * (1.f / D_OUT) - mean * mean;
    const float rstd = rsqrtf(var + LN_EPS);

    float* orow = out + (size_t)(rBase + row) * D_OUT;
#pragma unroll
    for (int j = 0; j < 8; ++j) {
        const int c = lane + 32 * j;                             // 128B-contig stores
        const float y = (vals[j] - mean) * rstd * gamma[c] + beta[c];
        orow[c] = fmaxf(y, 0.f);
    }
}

// -------------------------------------------------------------------------
extern "C" void kernel_launch(void* const* d_in, const int* in_sizes, int n_in,
                              void* d_out, int out_size, void* d_ws, size_t ws_size,
                              hipStream_t stream) {
    const float*     emb   = (const float*)d_in[0];
    const long long* idx   = (const long long*)d_in[1];   // int64 indices
    const float*     W     = (const float*)d_in[2];
    const float*     bias  = (const float*)d_in[3];
    const float*     gamma = (const float*)d_in[4];
    const float*     beta  = (const float*)d_in[5];
    float*           out   = (float*)d_out;
    float*           pooled = (float*)d_ws;               // 8192*128 f32 = 4 MB

    gather_maxpool_kernel<<<N_ROUTES / ROUTES_PER_BLOCK, 256, 0, stream>>>(
        emb, idx, pooled);
    gemm_ln_relu_kernel<<<N_ROUTES / 16, 512, 0, stream>>>(pooled, W, bias,
                                                           gamma, beta, out);
}